// GraphSAGE_16982300688532
// MI455X (gfx1250) — compile-verified
//
#include <hip/hip_runtime.h>
#include <math.h>
#include <stdint.h>

// ---------------------------------------------------------------------------
// GraphSAGE forward for MI455X (gfx1250, wave32, WMMA + async global->LDS)
// ---------------------------------------------------------------------------

typedef __attribute__((ext_vector_type(2))) float v2f;
typedef __attribute__((ext_vector_type(8))) float v8f;

#define N_NODES 50000
#define N_EDGES 800000
#define DIN     96
#define DMID    128
#define NGRAPH  64
#define GDIN    32
#define EPSV    1e-12f

// ---- workspace layout (float offsets) -------------------------------------
#define OFF_CNT    0
#define OFF_SSUM0  (N_NODES)
#define OFF_SSUM1  (OFF_SSUM0 + N_NODES * DIN)
#define OFF_GSUM   (OFF_SSUM1 + N_NODES * DMID)
#define OFF_GCNT   (OFF_GSUM + NGRAPH * DMID)
#define ZERO_FLTS  (OFF_GCNT + NGRAPH)
#define OFF_H1     (ZERO_FLTS)
#define OFF_H2     (OFF_H1 + N_NODES * DMID)

// ---------------------------------------------------------------------------
// Edge scatter: ssum[dst] += feat[src] (float4 chunks); cnt[dst] += 1 once.
// ---------------------------------------------------------------------------
__global__ void scatter_kernel(const float* __restrict__ feat,
                               const int* __restrict__ src,
                               const int* __restrict__ dst,
                               float* __restrict__ ssum,
                               float* __restrict__ cnt,  // may be null
                               int K) {
  int tid = blockIdx.x * blockDim.x + threadIdx.x;
  int chunks = K >> 2;
  int e = tid / chunks;
  if (e >= N_EDGES) return;
  int c = (tid - e * chunks) << 2;
  int s = src[e];
  int d = dst[e];
  const float4 v = *(const float4*)(feat + (long)s * K + c);
  float* p = ssum + (long)d * K + c;
  atomicAdd(p + 0, v.x);
  atomicAdd(p + 1, v.y);
  atomicAdd(p + 2, v.z);
  atomicAdd(p + 3, v.w);
  if (cnt && c == 0) atomicAdd(cnt + d, 1.0f);
}

// ---------------------------------------------------------------------------
// Fused SAGEConv GEMM:  out = (ssum/max(cnt,1)) @ Wl + root @ Wr + b
//
// Block = 16-row strip; 8 waves = 8 column tiles of the 128 output cols.
// Both A strips (ssum + root, 16 x K each) are staged into LDS ONCE per
// block via async global->LDS copies (ASYNCcnt), then all 8 waves feed
// v_wmma_f32_16x16x4_f32 from LDS (ds_load_b64) -- removes the 8x redundant,
// uncoalesced global A reads.
//
// Fragment layouts per ISA 7.12.2 (wave32):
//   A (16x4): M = lane&15; lanes 0-15 hold K={0,1}, lanes 16-31 K={2,3}
//   B (4x16): N = lane&15; lanes 0-15 hold K={0,1}, lanes 16-31 K={2,3}
//   C (16x16): VGPR v -> M=v (lanes 0-15) / M=v+8 (lanes 16-31)
// The 1/cnt mean-scale is applied per C-row in the epilogue (two
// accumulators), keeping the inner loop pure load+wmma.
// EXEC is all-1s in the WMMA section (N_NODES % 16 == 0, no guards).
// ---------------------------------------------------------------------------
__global__ void sage_gemm_kernel(const float* __restrict__ Assum,
                                 const float* __restrict__ cnt,
                                 const float* __restrict__ Aroot,
                                 const float* __restrict__ Wl,
                                 const float* __restrict__ Wr,
                                 const float* __restrict__ bias,
                                 float* __restrict__ out,
                                 int K) {
  __shared__ float sA[16 * DMID];  // ssum strip, 16 x K
  __shared__ float sR[16 * DMID];  // root strip, 16 x K

  // ---- cooperative async stage of both A strips into LDS ----
  const int kc = K >> 2;        // float4 chunks per row
  const int nchunk = 16 * kc;   // 384 (K=96) or 512 (K=128)
  for (int i = threadIdx.x; i < nchunk; i += 256) {
    const int r = i / kc;
    const int c4 = (i - r * kc) << 2;
    const long gidx = (long)(blockIdx.x * 16 + r) * K + c4;
    {
      const float* g = Assum + gidx;
      unsigned l = (unsigned)(uintptr_t)&sA[r * K + c4];
      asm volatile("global_load_async_to_lds_b128 %0, %1, off"
                   :: "v"(l), "v"(g) : "memory");
    }
    {
      const float* g = Aroot + gidx;
      unsigned l = (unsigned)(uintptr_t)&sR[r * K + c4];
      asm volatile("global_load_async_to_lds_b128 %0, %1, off"
                   :: "v"(l), "v"(g) : "memory");
    }
  }
  asm volatile("s_wait_asynccnt 0x0" ::: "memory");
  __syncthreads();

  const int lane  = threadIdx.x & 31;
  const int wave  = threadIdx.x >> 5;         // column tile 0..7
  const int m16   = lane & 15;
  const int khalf = (lane >> 4) << 1;         // 0 or 2
  const int col   = wave * 16 + m16;          // B/C column for this lane

  v8f accA = {};
  v8f accR = {};
#pragma unroll 4
  for (int k = 0; k < K; k += 4) {
    const int kk = k + khalf;
    v2f a = *(const v2f*)&sA[m16 * K + kk];
    v2f b;
    b.x = Wl[kk * DMID + col];
    b.y = Wl[(kk + 1) * DMID + col];
    accA = __builtin_amdgcn_wmma_f32_16x16x4_f32(false, a, false, b,
                                                 (short)0, accA, false, false);
  }
#pragma unroll 4
  for (int k = 0; k < K; k += 4) {
    const int kk = k + khalf;
    v2f a = *(const v2f*)&sR[m16 * K + kk];
    v2f b;
    b.x = Wr[kk * DMID + col];
    b.y = Wr[(kk + 1) * DMID + col];
    accR = __builtin_amdgcn_wmma_f32_16x16x4_f32(false, a, false, b,
                                                 (short)0, accR, false, false);
  }

  const float bv = bias[col];
  const int rbase = blockIdx.x * 16 + ((lane >> 4) << 3);
#pragma unroll
  for (int i = 0; i < 8; ++i) {
    const int r = rbase + i;
    const float invc = 1.0f / fmaxf(cnt[r], 1.0f);  // mean-scale per C-row
    out[(long)r * DMID + col] = accA[i] * invc + accR[i] + bv;
  }
}

// ---------------------------------------------------------------------------
// Row L2-normalize (+ optional ReLU). One wave per 128-float row.
// ---------------------------------------------------------------------------
__global__ void rownorm_kernel(float* __restrict__ h, int do_relu) {
  const int lane = threadIdx.x & 31;
  const int wave = threadIdx.x >> 5;
  const int row  = blockIdx.x * 8 + wave;
  if (row >= N_NODES) return;
  float4 v = *(const float4*)(h + (long)row * DMID + lane * 4);
  float ss = v.x * v.x + v.y * v.y + v.z * v.z + v.w * v.w;
#pragma unroll
  for (int m = 16; m >= 1; m >>= 1) ss += __shfl_xor(ss, m, 32);
  const float scale = 1.0f / fmaxf(sqrtf(ss), EPSV);
  v.x *= scale; v.y *= scale; v.z *= scale; v.w *= scale;
  if (do_relu) {
    v.x = fmaxf(v.x, 0.0f); v.y = fmaxf(v.y, 0.0f);
    v.z = fmaxf(v.z, 0.0f); v.w = fmaxf(v.w, 0.0f);
  }
  *(float4*)(h + (long)row * DMID + lane * 4) = v;
}

// ---------------------------------------------------------------------------
// Per-graph pooling: gsum[batch[n]] += h2[n], gcnt[batch[n]] += 1.
// ---------------------------------------------------------------------------
__global__ void pool_kernel(const float* __restrict__ h2,
                            const int* __restrict__ batch,
                            float* __restrict__ gsum,
                            float* __restrict__ gcnt) {
  int tid = blockIdx.x * blockDim.x + threadIdx.x;
  int n = tid >> 5;
  if (n >= N_NODES) return;
  int c = (tid & 31) << 2;
  int g = batch[n];
  const float4 v = *(const float4*)(h2 + (long)n * DMID + c);
  float* p = gsum + g * DMID + c;
  atomicAdd(p + 0, v.x);
  atomicAdd(p + 1, v.y);
  atomicAdd(p + 2, v.z);
  atomicAdd(p + 3, v.w);
  if (c == 0) atomicAdd(gcnt + g, 1.0f);
}

// ---------------------------------------------------------------------------
// Head: global_emb = relu(u@Wg+bg); fused = [gsum/gcnt, global_emb];
// out[g] = fused @ Wh + bh.  One wave per graph.
// ---------------------------------------------------------------------------
__global__ void head_kernel(const float* __restrict__ u,
                            const float* __restrict__ Wg,
                            const float* __restrict__ bg,
                            const float* __restrict__ Wh,
                            const float* __restrict__ bh,
                            const float* __restrict__ gsum,
                            const float* __restrict__ gcnt,
                            float* __restrict__ out) {
  const int lane = threadIdx.x & 31;
  const int wave = threadIdx.x >> 5;
  const int g    = blockIdx.x * 8 + wave;
  if (g >= NGRAPH) return;
  const float invc = 1.0f / fmaxf(gcnt[g], 1.0f);
  float partial = 0.0f;
#pragma unroll
  for (int j = 0; j < 4; ++j) {
    const int c = lane * 4 + j;
    float ge = bg[c];
#pragma unroll 8
    for (int k = 0; k < GDIN; ++k) ge += u[g * GDIN + k] * Wg[k * DMID + c];
    ge = fmaxf(ge, 0.0f);
    const float gm = gsum[g * DMID + c] * invc;
    partial += gm * Wh[c] + ge * Wh[DMID + c];
  }
#pragma unroll
  for (int m = 16; m >= 1; m >>= 1) partial += __shfl_xor(partial, m, 32);
  if (lane == 0) out[g] = partial + bh[0];
}

// ---------------------------------------------------------------------------
extern "C" void kernel_launch(void* const* d_in, const int* in_sizes, int n_in,
                              void* d_out, int out_size, void* d_ws, size_t ws_size,
                              hipStream_t stream) {
  const float* x    = (const float*)d_in[0];   // [N, 96]
  const int*   ei   = (const int*)d_in[1];     // [2, E]
  const float* u    = (const float*)d_in[2];   // [G, 32]
  const int*   bat  = (const int*)d_in[3];     // [N]
  const float* Wl0  = (const float*)d_in[4];   // [96, 128]
  const float* Wr0  = (const float*)d_in[5];   // [96, 128]
  const float* b0   = (const float*)d_in[6];   // [128]
  const float* Wl1  = (const float*)d_in[7];   // [128, 128]
  const float* Wr1  = (const float*)d_in[8];   // [128, 128]
  const float* b1   = (const float*)d_in[9];   // [128]
  const float* Wg   = (const float*)d_in[10];  // [32, 128]
  const float* bg   = (const float*)d_in[11];  // [128]
  const float* Wh   = (const float*)d_in[12];  // [256, 1]
  const float* bh   = (const float*)d_in[13];  // [1]
  float* out = (float*)d_out;

  float* ws    = (float*)d_ws;
  float* cnt   = ws + OFF_CNT;
  float* ssum0 = ws + OFF_SSUM0;
  float* ssum1 = ws + OFF_SSUM1;
  float* gsum  = ws + OFF_GSUM;
  float* gcnt  = ws + OFF_GCNT;
  float* h1    = ws + OFF_H1;
  float* h2    = ws + OFF_H2;

  const int* src = ei;            // edge_index[0]
  const int* dst = ei + N_EDGES;  // edge_index[1]

  // zero all accumulators in one shot (contiguous by layout; capture-safe)
  hipMemsetAsync(ws, 0, (size_t)ZERO_FLTS * sizeof(float), stream);

  // ---- layer 0 ----
  {
    int total = N_EDGES * (DIN / 4);
    scatter_kernel<<<(total + 255) / 256, 256, 0, stream>>>(x, src, dst, ssum0,
                                                            cnt, DIN);
    sage_gemm_kernel<<<N_NODES / 16, 256, 0, stream>>>(ssum0, cnt, x, Wl0, Wr0,
                                                       b0, h1, DIN);
    rownorm_kernel<<<N_NODES / 8, 256, 0, stream>>>(h1, 1);
  }

  // ---- layer 1 ----
  {
    int total = N_EDGES * (DMID / 4);
    scatter_kernel<<<(total + 255) / 256, 256, 0, stream>>>(h1, src, dst, ssum1,
                                                            nullptr, DMID);
    sage_gemm_kernel<<<N_NODES / 16, 256, 0, stream>>>(ssum1, cnt, h1, Wl1, Wr1,
                                                       b1, h2, DMID);
    rownorm_kernel<<<N_NODES / 8, 256, 0, stream>>>(h2, 0);
  }

  // ---- pool + head ----
  pool_kernel<<<(N_NODES * 32 + 255) / 256, 256, 0, stream>>>(h2, bat, gsum,
                                                              gcnt);
  head_kernel<<<NGRAPH / 8, 256, 0, stream>>>(u, Wg, bg, Wh, bh, gsum, gcnt,
                                              out);
}